// Rewire_GCN_23596550324874
// MI455X (gfx1250) — compile-verified
//
#include <hip/hip_runtime.h>

// ---------------------------------------------------------------------------
// GCN (2-layer) for gfx1250: f32 WMMA GEMMs + native f32 atomic aggregation.
// ---------------------------------------------------------------------------

#define NFEAT      512
#define NHID       128
#define NCLASS     40
#define NCLASS_PAD 48   // 3 x 16 N-tiles, zero padded

typedef __attribute__((ext_vector_type(2))) float v2f;
typedef __attribute__((ext_vector_type(8))) float v8f;

// Non-returning device-scope f32 atomic add. Forced to the native CDNA5
// GLOBAL_ATOMIC_ADD_F32 (no-return form, tracked by STOREcnt; S_ENDPGM's
// implicit wait-idle covers completion) so we never fall back to a CAS loop.
__device__ __forceinline__ void atomic_add_f32(float* p, float v) {
#if defined(__gfx1250__)
    asm volatile("global_atomic_add_f32 %0, %1, off scope:SCOPE_DEV"
                 :: "v"(p), "v"(v) : "memory");
#else
    __hip_atomic_fetch_add(p, v, __ATOMIC_RELAXED, __HIP_MEMORY_SCOPE_AGENT);
#endif
}

// -------------------------- degree / normalization -------------------------

__global__ void k_init_deg(float* __restrict__ deg, int n) {
    int i = blockIdx.x * blockDim.x + threadIdx.x;
    if (i < n) deg[i] = 1.0f;                       // self-loop weight
}

__global__ void k_accum_deg(const int* __restrict__ col,
                            const float* __restrict__ w,
                            float* __restrict__ deg, int E) {
    int e = blockIdx.x * blockDim.x + threadIdx.x;
    if (e < E) atomic_add_f32(&deg[col[e]], w[e]);
}

__global__ void k_dinv(float* __restrict__ deg, int n) {
    int i = blockIdx.x * blockDim.x + threadIdx.x;
    if (i < n) {
        float d = deg[i];
        deg[i] = (d > 0.0f) ? rsqrtf(d) : 0.0f;     // deg >= 1 always
    }
}

// ------------------------------- utilities ---------------------------------

__global__ void k_zero(float* __restrict__ p, long n) {
    long i = (long)blockIdx.x * blockDim.x + threadIdx.x;
    if (i < n) p[i] = 0.0f;
}

// Wt[n*K + k] = (n < Nin) ? W[k*Nin + n] : 0   for n in [0, Npad)
__global__ void k_transpose_pad(const float* __restrict__ W,
                                float* __restrict__ Wt,
                                int K, int Nin, int Npad) {
    int idx = blockIdx.x * blockDim.x + threadIdx.x;
    if (idx >= K * Npad) return;
    int n = idx / K;
    int k = idx - n * K;
    Wt[(long)n * K + k] = (n < Nin) ? W[(long)k * Nin + n] : 0.0f;
}

// ------------------------------ WMMA GEMM ----------------------------------
// C[M x ncols] = A[M x K] * B[K x ncols], B given transposed (Bt[n][k], ld K).
// One wave computes a 16-row M tile across NTILES 16-wide N tiles using
// V_WMMA_F32_16X16X4_F32 (full f32 precision).
//
// Fragment layouts (ISA 7.12.2, wave32):
//   A 16x4 : lane l -> row l&15, K = k0 + 2*(l>>4) + {0,1}   (float2 load)
//   B 4x16 : lane l -> col l&15, K = k0 + 2*(l>>4) + {0,1}   (float2 from Bt)
//   C 16x16: vgpr r -> row m0 + r + 8*(l>>4), col n0 + (l&15)

template <int K, int NTILES>
__global__ __launch_bounds__(256) void k_gemm_wmma(
        const float* __restrict__ A,
        const float* __restrict__ Bt,
        float* __restrict__ C,
        int M, int ncols) {
    const int wave = (blockIdx.x * blockDim.x + threadIdx.x) >> 5;
    const int lane = threadIdx.x & 31;
    const int m0   = wave * 16;
    if (m0 >= M) return;                 // wave-uniform: EXEC stays all-ones

    const int half = lane >> 4;          // 0 or 1
    const int l16  = lane & 15;

    v8f acc[NTILES] = {};

    const float* arow = A + (long)(m0 + l16) * K + 2 * half;

    for (int k0 = 0; k0 < K; k0 += 4) {
        v2f a = *(const v2f*)(arow + k0);
#pragma unroll
        for (int t = 0; t < NTILES; ++t) {
            v2f b = *(const v2f*)(Bt + (long)(t * 16 + l16) * K + k0 + 2 * half);
            acc[t] = __builtin_amdgcn_wmma_f32_16x16x4_f32(
                false, a, false, b, (short)0, acc[t], false, false);
        }
    }

#pragma unroll
    for (int t = 0; t < NTILES; ++t) {
        int n = t * 16 + l16;
        if (n < ncols) {
#pragma unroll
            for (int r = 0; r < 8; ++r) {
                int m = m0 + r + 8 * half;
                C[(long)m * ncols + n] = acc[t][r];
            }
        }
    }
}

// ---------------------------- edge aggregation -----------------------------
// out[col[e]][:] += dinv[row]*w*dinv[col] * h[row[e]][:]
// LPE lanes per edge, each active lane handles 4 contiguous dims (float4).

template <int DIM, int LPE>
__global__ void k_aggregate(const int* __restrict__ row,
                            const int* __restrict__ col,
                            const float* __restrict__ w,
                            const float* __restrict__ dinv,
                            const float* __restrict__ h,
                            float* __restrict__ out, int E) {
    long gid = (long)blockIdx.x * blockDim.x + threadIdx.x;
    int e = (int)(gid >> (LPE == 32 ? 5 : 4));
    int l = (int)(gid & (LPE - 1));
    if (e >= E) return;
    int d = l * 4;
    if (d >= DIM) return;

    int r = row[e];
    int c = col[e];
    float nrm = dinv[r] * w[e] * dinv[c];

    float4 hv = *(const float4*)(h + (long)r * DIM + d);
    float* op = out + (long)c * DIM + d;
    atomic_add_f32(op + 0, nrm * hv.x);
    atomic_add_f32(op + 1, nrm * hv.y);
    atomic_add_f32(op + 2, nrm * hv.z);
    atomic_add_f32(op + 3, nrm * hv.w);
}

// agg1 <- relu(agg1 + dinv^2 * h1 + b1)   (in place, becomes layer-2 input)
__global__ void k_finalize1(float* __restrict__ agg,
                            const float* __restrict__ h1,
                            const float* __restrict__ dinv,
                            const float* __restrict__ b, int nNodes) {
    long gid = (long)blockIdx.x * blockDim.x + threadIdx.x;
    int i = (int)(gid >> 5);
    int d = (int)(gid & 31) * 4;
    if (i >= nNodes) return;
    float di = dinv[i];
    float s = di * di;
    float4 a  = *(float4*)(agg + (long)i * NHID + d);
    float4 hv = *(const float4*)(h1 + (long)i * NHID + d);
    float4 bv = *(const float4*)(b + d);
    a.x = fmaxf(fmaf(s, hv.x, a.x) + bv.x, 0.0f);
    a.y = fmaxf(fmaf(s, hv.y, a.y) + bv.y, 0.0f);
    a.z = fmaxf(fmaf(s, hv.z, a.z) + bv.z, 0.0f);
    a.w = fmaxf(fmaf(s, hv.w, a.w) + bv.w, 0.0f);
    *(float4*)(agg + (long)i * NHID + d) = a;
}

// out <- out + dinv^2 * g + b2   (out already holds edge-aggregated sums)
__global__ void k_finalize2(float* __restrict__ out,
                            const float* __restrict__ g,
                            const float* __restrict__ dinv,
                            const float* __restrict__ b, int nNodes) {
    long gid = (long)blockIdx.x * blockDim.x + threadIdx.x;
    int i = (int)(gid / 10);
    int d = (int)(gid - (long)i * 10) * 4;
    if (i >= nNodes) return;
    float di = dinv[i];
    float s = di * di;
    float4 o  = *(float4*)(out + (long)i * NCLASS + d);
    float4 gv = *(const float4*)(g + (long)i * NCLASS + d);
    float4 bv = *(const float4*)(b + d);
    o.x = fmaf(s, gv.x, o.x) + bv.x;
    o.y = fmaf(s, gv.y, o.y) + bv.y;
    o.z = fmaf(s, gv.z, o.z) + bv.z;
    o.w = fmaf(s, gv.w, o.w) + bv.w;
    *(float4*)(out + (long)i * NCLASS + d) = o;
}

// ------------------------------- launcher ----------------------------------

extern "C" void kernel_launch(void* const* d_in, const int* in_sizes, int n_in,
                              void* d_out, int out_size, void* d_ws, size_t ws_size,
                              hipStream_t stream) {
    const float* x  = (const float*)d_in[0];
    const int*   ei = (const int*)d_in[1];
    const float* ew = (const float*)d_in[2];
    const float* W1 = (const float*)d_in[3];
    const float* b1 = (const float*)d_in[4];
    const float* W2 = (const float*)d_in[5];
    const float* b2 = (const float*)d_in[6];
    float* out = (float*)d_out;

    const int nNodes = in_sizes[0] / NFEAT;
    const int E      = in_sizes[1] / 2;
    const int* row = ei;        // edge_index[0]
    const int* col = ei + E;    // edge_index[1]

    // workspace carve-out (floats)
    float* ws   = (float*)d_ws;
    float* deg  = ws;                                   // nNodes (becomes dinv)
    float* w1t  = deg + nNodes;                         // NFEAT*NHID
    float* w2t  = w1t + (long)NFEAT * NHID;             // NHID*NCLASS_PAD
    float* h1   = w2t + (long)NHID * NCLASS_PAD;        // nNodes*NHID
    float* agg1 = h1 + (long)nNodes * NHID;             // nNodes*NHID
    float* g2   = agg1 + (long)nNodes * NHID;           // nNodes*NCLASS

    const int T = 256;
    auto nb = [](long n, int t) { return (unsigned)((n + t - 1) / t); };

    // --- normalization ---
    k_init_deg<<<nb(nNodes, T), T, 0, stream>>>(deg, nNodes);
    k_accum_deg<<<nb(E, T), T, 0, stream>>>(col, ew, deg, E);
    k_dinv<<<nb(nNodes, T), T, 0, stream>>>(deg, nNodes);

    // --- weight transposes (B fragments become contiguous float2) ---
    k_transpose_pad<<<nb((long)NFEAT * NHID, T), T, 0, stream>>>(W1, w1t, NFEAT, NHID, NHID);
    k_transpose_pad<<<nb((long)NHID * NCLASS_PAD, T), T, 0, stream>>>(W2, w2t, NHID, NCLASS, NCLASS_PAD);

    // --- zero accumulators ---
    k_zero<<<nb((long)nNodes * NHID, T), T, 0, stream>>>(agg1, (long)nNodes * NHID);
    k_zero<<<nb((long)out_size, T), T, 0, stream>>>(out, (long)out_size);

    // --- layer 1: h1 = x @ W1 (WMMA f32) ---
    {
        long waves = ((long)nNodes + 15) / 16;
        k_gemm_wmma<NFEAT, NHID / 16><<<nb(waves * 32, T), T, 0, stream>>>(
            x, w1t, h1, nNodes, NHID);
    }
    // aggregate over edges, then self-loop + bias + relu (into agg1)
    k_aggregate<NHID, 32><<<nb((long)E * 32, T), T, 0, stream>>>(
        row, col, ew, deg, h1, agg1, E);
    k_finalize1<<<nb((long)nNodes * 32, T), T, 0, stream>>>(agg1, h1, deg, b1, nNodes);

    // --- layer 2: g2 = agg1 @ W2 (WMMA f32, N padded to 48) ---
    {
        long waves = ((long)nNodes + 15) / 16;
        k_gemm_wmma<NHID, NCLASS_PAD / 16><<<nb(waves * 32, T), T, 0, stream>>>(
            agg1, w2t, g2, nNodes, NCLASS);
    }
    k_aggregate<NCLASS, 16><<<nb((long)E * 16, T), T, 0, stream>>>(
        row, col, ew, deg, g2, out, E);
    k_finalize2<<<nb((long)nNodes * 10, T), T, 0, stream>>>(out, g2, deg, b2, nNodes);
}